// AttentionFlowLayer_87162066305636
// MI455X (gfx1250) — compile-verified
//
#include <hip/hip_runtime.h>
#include <hip/hip_bf16.h>
#include <math.h>

// ---------------------------------------------------------------------------
// AttentionFlow layer for MI455X (gfx1250), f32 precision throughout.
// All matmuls use V_WMMA_F32_16X16X4_F32 (wave32, A=16x4, B=4x16, C/D=16x16).
// WMMA kernels run 2 waves per workgroup sharing staged LDS so both
// workgroups/WGP together fill all 4 SIMD32s.  Output (164 MB, write-once)
// uses non-temporal stores to keep c/q/S resident in the 192 MB L2.
// ---------------------------------------------------------------------------

#define B_    32
#define CLEN_ 400
#define QLEN_ 50
#define D_    800
#define QP_   64      // padded j extent (softmax weights zero-filled 50..63)
#define OUTW_ (4 * D_)

typedef __attribute__((ext_vector_type(2))) float v2f;
typedef __attribute__((ext_vector_type(8))) float v8f;

static __device__ __forceinline__ v8f wmma4(v2f a, v2f b, v8f c) {
  // D = A(16x4,f32) * B(4x16,f32) + C(16x16,f32)
  return __builtin_amdgcn_wmma_f32_16x16x4_f32(
      /*neg_a=*/false, a, /*neg_b=*/false, b,
      /*c_mod=*/(short)0, c, /*reuse_a=*/false, /*reuse_b=*/false);
}

static __device__ __forceinline__ void ntstore(float v, float* p) {
  __builtin_nontemporal_store(v, p);
}

static __device__ __forceinline__ float wave_sum(float v) {
  for (int off = 16; off; off >>= 1) v += __shfl_xor(v, off, 32);
  return v;
}
static __device__ __forceinline__ float wave_max(float v) {
  for (int off = 16; off; off >>= 1) v = fmaxf(v, __shfl_xor(v, off, 32));
  return v;
}

// ---------------------------------------------------------------------------
// K0: rowdot[b,i] = c[b,i,:].w_c ; coldot[b,j] = q[b,j,:].w_q
// One wave per row; lane-strided coalesced loads + shfl reduction.
// ---------------------------------------------------------------------------
__global__ void k_dots(const float* __restrict__ c, const float* __restrict__ q,
                       const float* __restrict__ W0,
                       float* __restrict__ rowdot, float* __restrict__ coldot) {
  const int lane = threadIdx.x & 31;
  const int g = blockIdx.x * 8 + (threadIdx.x >> 5);
  const int NC = B_ * CLEN_;            // 12800 c-rows
  const int NQ = B_ * QLEN_;            // 1600 q-rows
  if (g < NC) {
    const float* r = c + (long)g * D_;
    float s = 0.f;
    for (int d = lane; d < D_; d += 32) s += r[d] * W0[d];
    s = wave_sum(s);
    if (lane == 0) rowdot[g] = s;
  } else if (g < NC + NQ) {
    const int h = g - NC;
    const float* r = q + (long)h * D_;
    float s = 0.f;
    for (int d = lane; d < D_; d += 32) s += r[d] * W0[D_ + d];
    s = wave_sum(s);
    if (lane == 0) coldot[h] = s;
  }
}

// ---------------------------------------------------------------------------
// K1: S[b,i,j] = rowdot + coldot + sum_d (c*w_cq)[b,i,d] * q[b,j,d]
// One workgroup (2 waves) per 16-row i-tile.  The cw tile is staged once and
// shared; each wave keeps a private q j-tile buffer and covers 2 j-tiles.
// LDS row stride 804 floats -> conflict-free lane access.  WMMA over K=800.
// ---------------------------------------------------------------------------
#define LSTRIDE 804
__global__ __launch_bounds__(64) void k_s(const float* __restrict__ c,
                                          const float* __restrict__ q,
                                          const float* __restrict__ W0,
                                          const float* __restrict__ rowdot,
                                          const float* __restrict__ coldot,
                                          float* __restrict__ S) {
  extern __shared__ float smem[];
  float* cwL = smem;                                   // [16][804] shared
  const int wid  = threadIdx.x >> 5;
  const int lane = threadIdx.x & 31;
  float* qL = smem + 16 * LSTRIDE + wid * 16 * LSTRIDE; // [16][804] per-wave

  const int m16  = lane & 15;        // A row / B col
  const int kb   = (lane >> 4) * 2;  // K base within fragment
  const int hi   = lane >> 4;

  const int b  = blockIdx.x / (CLEN_ / 16);
  const int i0 = (blockIdx.x % (CLEN_ / 16)) * 16;

  const float* wcq = W0 + 2 * D_;
  const float* ctile = c + ((long)b * CLEN_ + i0) * D_;

  // Cooperative stage of cw = c * w_cq (waves take 8 rows each)
  for (int r = wid * 8; r < wid * 8 + 8; ++r) {
    const float4* src = (const float4*)(ctile + (long)r * D_);
    const float4* w4  = (const float4*)wcq;
    float4* dst = (float4*)(cwL + r * LSTRIDE);
    for (int t = lane; t < D_ / 4; t += 32) {
      float4 v = src[t], w = w4[t];
      v.x *= w.x; v.y *= w.y; v.z *= w.z; v.w *= w.w;
      dst[t] = v;
    }
  }
  __syncthreads();

  for (int jt = wid * 2; jt < wid * 2 + 2; ++jt) {
    // Stage this wave's q j-tile (rows >= QLEN zero-filled); private buffer,
    // same-wave LDS ordering makes a barrier unnecessary.
    for (int r = 0; r < 16; ++r) {
      const int jr = jt * 16 + r;
      float4* dst = (float4*)(qL + r * LSTRIDE);
      if (jr < QLEN_) {
        const float4* src = (const float4*)(q + ((long)b * QLEN_ + jr) * D_);
        for (int t = lane; t < D_ / 4; t += 32) dst[t] = src[t];
      } else {
        const float4 z = {0.f, 0.f, 0.f, 0.f};
        for (int t = lane; t < D_ / 4; t += 32) dst[t] = z;
      }
    }

    v8f acc = {};
    for (int dk = 0; dk < D_; dk += 4) {
      v2f a, bb;
      a.x  = cwL[m16 * LSTRIDE + dk + kb];
      a.y  = cwL[m16 * LSTRIDE + dk + kb + 1];
      bb.x = qL [m16 * LSTRIDE + dk + kb];      // B[d][j] = q[j][d]
      bb.y = qL [m16 * LSTRIDE + dk + kb + 1];
      acc = wmma4(a, bb, acc);
    }

    #pragma unroll
    for (int r = 0; r < 8; ++r) {
      const int gi = i0 + r + hi * 8;
      const int gj = jt * 16 + m16;
      float v = acc[r] + rowdot[b * CLEN_ + gi];
      if (gj < QLEN_) v += coldot[b * QLEN_ + gj];
      S[((long)b * CLEN_ + gi) * QP_ + gj] = v;
    }
  }
}

// ---------------------------------------------------------------------------
// K2: softmax over j (length 50) per (b,i) row -> S2 (zero-padded to 64)
// ---------------------------------------------------------------------------
__global__ void k_softmax_j(const float* __restrict__ S, float* __restrict__ S2) {
  const int lane = threadIdx.x & 31;
  const int g = blockIdx.x * 8 + (threadIdx.x >> 5);       // row id, < 12800
  const float* row = S + (long)g * QP_;
  const int j2 = lane + 32;
  const float x0 = row[lane];
  const float x1 = (j2 < QLEN_) ? row[j2] : -INFINITY;
  const float m = wave_max(fmaxf(x0, x1));
  const float e0 = __expf(x0 - m);
  const float e1 = (j2 < QLEN_) ? __expf(x1 - m) : 0.f;
  const float inv = 1.f / wave_sum(e0 + e1);
  float* out = S2 + (long)g * QP_;
  out[lane] = e0 * inv;
  out[j2]   = (j2 < QLEN_) ? e1 * inv : 0.f;
}

// ---------------------------------------------------------------------------
// K3: softmax over i (length 400) per (b,j) column -> S1 (j>=50 columns = 0)
// One wave per (b, padded j).
// ---------------------------------------------------------------------------
__global__ void k_softmax_i(const float* __restrict__ S, float* __restrict__ S1) {
  const int lane = threadIdx.x & 31;
  const int g = blockIdx.x * 8 + (threadIdx.x >> 5);       // < 32*64
  const int b = g >> 6, j = g & 63;
  float* dst = S1 + ((long)b * CLEN_) * QP_ + j;
  if (j >= QLEN_) {
    for (int i = lane; i < CLEN_; i += 32) dst[(long)i * QP_] = 0.f;
    return;
  }
  const float* src = S + ((long)b * CLEN_) * QP_ + j;
  float xs[13];
  float m = -INFINITY;
  #pragma unroll
  for (int t = 0; t < 13; ++t) {
    const int i = t * 32 + lane;
    xs[t] = (i < CLEN_) ? src[(long)i * QP_] : -INFINITY;
    m = fmaxf(m, xs[t]);
  }
  m = wave_max(m);
  float s = 0.f;
  #pragma unroll
  for (int t = 0; t < 13; ++t) {
    xs[t] = (t * 32 + lane < CLEN_) ? __expf(xs[t] - m) : 0.f;
    s += xs[t];
  }
  const float inv = 1.f / wave_sum(s);
  #pragma unroll
  for (int t = 0; t < 13; ++t) {
    const int i = t * 32 + lane;
    if (i < CLEN_) dst[(long)i * QP_] = xs[t] * inv;
  }
}

// ---------------------------------------------------------------------------
// K4a: A = S1 @ q  (16-row tile per workgroup, 2 waves split the d-tiles),
// fused non-temporal write of out slots {c, A, c*A}.
// ---------------------------------------------------------------------------
__global__ __launch_bounds__(64) void k_a(const float* __restrict__ c,
                                          const float* __restrict__ q,
                                          const float* __restrict__ S1,
                                          float* __restrict__ out) {
  extern __shared__ float smem[];
  float* s1L = smem;                 // [16][64], pad already zero

  const int wid  = threadIdx.x >> 5;
  const int lane = threadIdx.x & 31;
  const int m16 = lane & 15, kb = (lane >> 4) * 2, hi = lane >> 4;
  const int b  = blockIdx.x / (CLEN_ / 16);
  const int i0 = (blockIdx.x % (CLEN_ / 16)) * 16;

  const float4* src = (const float4*)(S1 + ((long)b * CLEN_ + i0) * QP_);
  for (int t = threadIdx.x; t < 16 * QP_ / 4; t += 64) ((float4*)s1L)[t] = src[t];
  __syncthreads();

  const float* qb = q + (long)b * QLEN_ * D_;
  for (int dt = wid; dt < D_ / 16; dt += 2) {
    const int d0 = dt * 16;
    v8f acc = {};
    for (int js = 0; js < 13; ++js) {            // K = 52 (S1 zero at 50,51)
      const int jj = js * 4 + kb;
      v2f a, bb;
      a.x = s1L[m16 * QP_ + jj];
      a.y = s1L[m16 * QP_ + jj + 1];
      const int j0c = (jj     < QLEN_) ? jj     : QLEN_ - 1;
      const int j1c = (jj + 1 < QLEN_) ? jj + 1 : QLEN_ - 1;
      float bx = qb[(long)j0c * D_ + d0 + m16];
      float by = qb[(long)j1c * D_ + d0 + m16];
      bb.x = (jj     < QLEN_) ? bx : 0.f;
      bb.y = (jj + 1 < QLEN_) ? by : 0.f;
      acc = wmma4(a, bb, acc);
    }
    #pragma unroll
    for (int r = 0; r < 8; ++r) {
      const int gi = i0 + r + hi * 8;
      const int gc = d0 + m16;
      const float cv = c[((long)b * CLEN_ + gi) * D_ + gc];
      float* ob = out + ((long)b * CLEN_ + gi) * OUTW_;
      const float av = acc[r];
      ntstore(cv,      ob + gc);           // slot 0: c
      ntstore(av,      ob + D_ + gc);      // slot 1: A
      ntstore(cv * av, ob + 2 * D_ + gc);  // slot 2: c*A
    }
  }
}

// ---------------------------------------------------------------------------
// K4b: M_tile(16x400) = S1_tile @ S2[b]^T, then Bv = M_tile @ c[b];
// fused non-temporal write of out slot 3 = c * Bv.
// 2 waves per workgroup share {S1 tile, S2[b], M tile} in LDS and split the
// kt (M) and dt (Bv) loops; one barrier between M-write and Bv-read.
// ---------------------------------------------------------------------------
#define S2STRIDE 68
#define MSTRIDE  404
__global__ __launch_bounds__(64) void k_mb(const float* __restrict__ c,
                                           const float* __restrict__ S1,
                                           const float* __restrict__ S2,
                                           float* __restrict__ out) {
  extern __shared__ float smem[];
  float* s1L = smem;                            // [16][64]
  float* s2L = s1L + 16 * QP_;                  // [400][68]
  float* mL  = s2L + CLEN_ * S2STRIDE;          // [16][404]

  const int wid  = threadIdx.x >> 5;
  const int lane = threadIdx.x & 31;
  const int m16 = lane & 15, kb = (lane >> 4) * 2, hi = lane >> 4;
  const int b  = blockIdx.x / (CLEN_ / 16);
  const int i0 = (blockIdx.x % (CLEN_ / 16)) * 16;

  const float4* src1 = (const float4*)(S1 + ((long)b * CLEN_ + i0) * QP_);
  for (int t = threadIdx.x; t < 16 * QP_ / 4; t += 64) ((float4*)s1L)[t] = src1[t];

  const float* src2 = S2 + (long)b * CLEN_ * QP_;
  for (int t = threadIdx.x; t < CLEN_ * QP_; t += 64) {
    const int r = t >> 6, jj = t & 63;
    s2L[r * S2STRIDE + jj] = src2[t];
  }
  __syncthreads();

  // M[i,k] = sum_j S1[i,j] * S2[k,j]   (B fragment: B[j][k] = S2[k][j])
  for (int kt = wid; kt < CLEN_ / 16; kt += 2) {
    v8f acc = {};
    for (int js = 0; js < 13; ++js) {
      const int jj = js * 4 + kb;
      v2f a, bb;
      a.x  = s1L[m16 * QP_ + jj];
      a.y  = s1L[m16 * QP_ + jj + 1];
      bb.x = s2L[(kt * 16 + m16) * S2STRIDE + jj];
      bb.y = s2L[(kt * 16 + m16) * S2STRIDE + jj + 1];
      acc = wmma4(a, bb, acc);
    }
    #pragma unroll
    for (int r = 0; r < 8; ++r)
      mL[(r + hi * 8) * MSTRIDE + kt * 16 + m16] = acc[r];
  }
  __syncthreads();

  // Bv = M_tile @ c[b]  (K = 400; B fragments are coalesced global loads)
  const float* cb = c + (long)b * CLEN_ * D_;
  for (int dt = wid; dt < D_ / 16; dt += 2) {
    const int d0 = dt * 16;
    if (dt + 2 < D_ / 16) __builtin_prefetch(cb + d0 + 32, 0, 0);
    v8f acc = {};
    for (int ks = 0; ks < CLEN_ / 4; ++ks) {
      const int kk = ks * 4 + kb;
      v2f a, bb;
      a.x  = mL[m16 * MSTRIDE + kk];
      a.y  = mL[m16 * MSTRIDE + kk + 1];
      bb.x = cb[(long)kk * D_ + d0 + m16];
      bb.y = cb[(long)(kk + 1) * D_ + d0 + m16];
      acc = wmma4(a, bb, acc);
    }
    #pragma unroll
    for (int r = 0; r < 8; ++r) {
      const int gi = i0 + r + hi * 8;
      const int gc = d0 + m16;
      const float cv = cb[(long)gi * D_ + gc];
      ntstore(cv * acc[r],
              out + ((long)b * CLEN_ + gi) * OUTW_ + 3 * D_ + gc); // c*Bv
    }
  }
}

// ---------------------------------------------------------------------------
extern "C" void kernel_launch(void* const* d_in, const int* in_sizes, int n_in,
                              void* d_out, int out_size, void* d_ws, size_t ws_size,
                              hipStream_t stream) {
  const float* c  = (const float*)d_in[0];
  const float* q  = (const float*)d_in[1];
  const float* W0 = (const float*)d_in[2];
  float* out = (float*)d_out;

  // Workspace partition (floats): rowdot | coldot | S | S1 | S2  (~9.9 MB)
  float* ws     = (float*)d_ws;
  float* rowdot = ws;                                  // 12800
  float* coldot = rowdot + B_ * CLEN_;                 // 1600
  float* S      = coldot + B_ * QLEN_;                 // 32*400*64
  float* S1     = S  + (long)B_ * CLEN_ * QP_;
  float* S2     = S1 + (long)B_ * CLEN_ * QP_;

  const int tiles = B_ * (CLEN_ / 16);                 // 800 tile-workgroups

  // K0: dot products (14400 waves)
  k_dots<<<(B_ * (CLEN_ + QLEN_)) / 8, 256, 0, stream>>>(c, q, W0, rowdot, coldot);

  // K1: S via WMMA (shared cw tile + 2 per-wave q tiles: 48*804 floats LDS)
  k_s<<<tiles, 64, (size_t)48 * LSTRIDE * sizeof(float), stream>>>(
      c, q, W0, rowdot, coldot, S);

  // K2/K3: softmaxes
  k_softmax_j<<<(B_ * CLEN_) / 8, 256, 0, stream>>>(S, S2);
  k_softmax_i<<<(B_ * QP_) / 8, 256, 0, stream>>>(S, S1);

  // K4a: A + out slots {c, A, c*A}
  k_a<<<tiles, 64, 16 * QP_ * sizeof(float), stream>>>(c, q, S1, out);

  // K4b: M + Bv + out slot {c*Bv}
  const size_t mb_lds =
      (size_t)(16 * QP_ + CLEN_ * S2STRIDE + 16 * MSTRIDE) * sizeof(float);
  k_mb<<<tiles, 64, mb_lds, stream>>>(c, S1, S2, out);
}